// CompetitiveNetwork_2_30597347016792
// MI455X (gfx1250) — compile-verified
//
#include <hip/hip_runtime.h>
#include <cstddef>

// ---------------------------------------------------------------------------
// Problem constants (match reference)
// ---------------------------------------------------------------------------
#define NAB   768
#define N2    (768 * 768)        // 589824
#define NY    512
#define TOLF  1e-6f
#define MAX_IT 500

// Solver grid: 96 blocks x 256 thr = 768 wave32 waves (one per K row)
#define SBLK     96
#define STHREADS 256
#define CONVQ    (N2 / (SBLK * STHREADS))   // 24 elements per thread

// GEMV grid: 512 blocks x 256 thr = 4096 waves = 32 y-tiles x 128 k-slices
#define GBLK     512
#define GTHREADS 256
#define KS       128
#define KLEN     (N2 / KS)       // 4608
#define IPERS    (KLEN / NAB)    // 6 K-rows per slice

// ws layout (float indices). ctrl words are u32.
#define WS_CTRL 0                // [0]=bar_cnt [1]=bar_gen [2]=md0 [3]=md1
#define WS_AF0  16
#define WS_BF0  (16 + 2 * NAB)   // AF0/AF1 contiguous, then BF0/BF1
#define WS_AFS  (16 + 4 * NAB)
#define WS_BFS  (16 + 5 * NAB)
#define WS_PART (16 + 6 * NAB)   // KS*NY floats of partials

typedef __attribute__((ext_vector_type(2))) float v2f;
typedef __attribute__((ext_vector_type(8))) float v8f;

// ---------------------------------------------------------------------------
// Device-wide sense barrier (persistent kernel; 96 co-resident blocks)
// ---------------------------------------------------------------------------
__device__ __forceinline__ void grid_barrier(unsigned* cnt, unsigned* gen) {
  __threadfence();
  __syncthreads();
  if (threadIdx.x == 0) {
    unsigned g = __hip_atomic_load(gen, __ATOMIC_RELAXED, __HIP_MEMORY_SCOPE_AGENT);
    unsigned a = __hip_atomic_fetch_add(cnt, 1u, __ATOMIC_ACQ_REL, __HIP_MEMORY_SCOPE_AGENT);
    if (a == SBLK - 1) {
      __hip_atomic_store(cnt, 0u, __ATOMIC_RELAXED, __HIP_MEMORY_SCOPE_AGENT);
      __hip_atomic_fetch_add(gen, 1u, __ATOMIC_ACQ_REL, __HIP_MEMORY_SCOPE_AGENT);
    } else {
      while (__hip_atomic_load(gen, __ATOMIC_ACQUIRE, __HIP_MEMORY_SCOPE_AGENT) == g)
        __builtin_amdgcn_s_sleep(1);
    }
  }
  __syncthreads();
}

// ---------------------------------------------------------------------------
// Init: barrier/maxdiff control words, AF0=AT, BF0=BT  (one 1024-thr block)
// ---------------------------------------------------------------------------
__global__ void init_kernel(const float* __restrict__ AT,
                            const float* __restrict__ BT,
                            float* __restrict__ ws) {
  int t = threadIdx.x;
  if (t < 4) ((unsigned*)ws)[t] = 0u;
  if (t < NAB) {
    ws[WS_AF0 + t] = AT[t];
    ws[WS_BF0 + t] = BT[t];
  }
}

// ---------------------------------------------------------------------------
// Persistent fixed-point solver + one extra (differentiable) iterate.
// Writes converged-then-iterated AF* / BF* to WS_AFS / WS_BFS.
// ---------------------------------------------------------------------------
__global__ void __launch_bounds__(STHREADS)
solve_kernel(const float* __restrict__ AT, const float* __restrict__ BT,
             const float* __restrict__ K, float* __restrict__ ws) {
  unsigned* ctrl = (unsigned*)ws;
  const int wv   = blockIdx.x * (STHREADS / 32) + (threadIdx.x >> 5); // 0..767
  const int lane = threadIdx.x & 31;

  int par = 0;
  int it  = 0;
  for (;;) {
    const float* AFo = ws + WS_AF0 + par * NAB;
    const float* BFo = ws + WS_BF0 + par * NAB;
    float*       AFn = ws + WS_AF0 + (par ^ 1) * NAB;
    float*       BFn = ws + WS_BF0 + (par ^ 1) * NAB;

    // ---- step A: AFn[r] = AT[r] / (1 + K[r,:] . BFo)   (one wave per row)
    {
      const float* Kr = K + (size_t)wv * NAB;
      float s0 = 0.f, s1 = 0.f, s2 = 0.f, s3 = 0.f;
#pragma unroll
      for (int t = 0; t < NAB; t += 128) {
        s0 += Kr[t + lane      ] * BFo[t + lane      ];
        s1 += Kr[t + lane + 32 ] * BFo[t + lane + 32 ];
        s2 += Kr[t + lane + 64 ] * BFo[t + lane + 64 ];
        s3 += Kr[t + lane + 96 ] * BFo[t + lane + 96 ];
      }
      float s = (s0 + s1) + (s2 + s3);
      for (int off = 16; off; off >>= 1) s += __shfl_xor(s, off, 32);
      if (lane == 0) AFn[wv] = AT[wv] / (1.0f + s);
    }
    grid_barrier(ctrl, ctrl + 1);

    // ---- step B: BFn[j] = BT[j] / (1 + K[:,j] . AFn)   (24 column waves)
    if (wv < NAB / 32) {
      const int j = wv * 32 + lane;
      float a0 = 0.f, a1 = 0.f, a2 = 0.f, a3 = 0.f;
      for (int i = 0; i < NAB; i += 4) {
        a0 += K[(size_t)(i    ) * NAB + j] * AFn[i    ];
        a1 += K[(size_t)(i + 1) * NAB + j] * AFn[i + 1];
        a2 += K[(size_t)(i + 2) * NAB + j] * AFn[i + 2];
        a3 += K[(size_t)(i + 3) * NAB + j] * AFn[i + 3];
      }
      BFn[j] = BT[j] / (1.0f + ((a0 + a1) + (a2 + a3)));
    }
    grid_barrier(ctrl, ctrl + 1);

    // ---- convergence: max_ij |K_ij (AFn_i BFn_j - AFo_i BFo_j)|
    const int s_idx = it & 1;
    {
      float mx = 0.f;
      unsigned tg = blockIdx.x * STHREADS + threadIdx.x;
#pragma unroll
      for (int q = 0; q < CONVQ; ++q) {
        unsigned p = tg + (unsigned)q * (SBLK * STHREADS);
        unsigned i = p / NAB;
        unsigned j = p - i * NAB;
        float d = fabsf(K[p] * (AFn[i] * BFn[j] - AFo[i] * BFo[j]));
        mx = fmaxf(mx, d);
      }
      for (int off = 16; off; off >>= 1) mx = fmaxf(mx, __shfl_xor(mx, off, 32));
      if (blockIdx.x == 0 && threadIdx.x == 0)   // pre-zero the *other* slot
        __hip_atomic_store(&ctrl[2 + (s_idx ^ 1)], 0u, __ATOMIC_RELAXED,
                           __HIP_MEMORY_SCOPE_AGENT);
      if (lane == 0) atomicMax(&ctrl[2 + s_idx], __float_as_uint(mx));
    }
    grid_barrier(ctrl, ctrl + 1);

    float gm = __uint_as_float(__hip_atomic_load(&ctrl[2 + s_idx], __ATOMIC_ACQUIRE,
                                                 __HIP_MEMORY_SCOPE_AGENT));
    ++it;
    par ^= 1;
    if (gm <= TOLF || it >= MAX_IT) break;
  }

  // ---- one extra differentiable iterate -> AF*, BF*
  {
    const float* BFo = ws + WS_BF0 + par * NAB;
    float*       AFs = ws + WS_AFS;
    float*       BFs = ws + WS_BFS;
    const float* Kr  = K + (size_t)wv * NAB;
    float s0 = 0.f, s1 = 0.f, s2 = 0.f, s3 = 0.f;
#pragma unroll
    for (int t = 0; t < NAB; t += 128) {
      s0 += Kr[t + lane      ] * BFo[t + lane      ];
      s1 += Kr[t + lane + 32 ] * BFo[t + lane + 32 ];
      s2 += Kr[t + lane + 64 ] * BFo[t + lane + 64 ];
      s3 += Kr[t + lane + 96 ] * BFo[t + lane + 96 ];
    }
    float s = (s0 + s1) + (s2 + s3);
    for (int off = 16; off; off >>= 1) s += __shfl_xor(s, off, 32);
    if (lane == 0) AFs[wv] = AT[wv] / (1.0f + s);
    grid_barrier(ctrl, ctrl + 1);
    if (wv < NAB / 32) {
      const int j = wv * 32 + lane;
      float a0 = 0.f, a1 = 0.f, a2 = 0.f, a3 = 0.f;
      for (int i = 0; i < NAB; i += 4) {
        a0 += K[(size_t)(i    ) * NAB + j] * AFs[i    ];
        a1 += K[(size_t)(i + 1) * NAB + j] * AFs[i + 1];
        a2 += K[(size_t)(i + 2) * NAB + j] * AFs[i + 2];
        a3 += K[(size_t)(i + 3) * NAB + j] * AFs[i + 3];
      }
      BFs[j] = BT[j] / (1.0f + ((a0 + a1) + (a2 + a3)));
    }
  }
}

// ---------------------------------------------------------------------------
// Y_partials = W @ vec(C), C_ij = K_ij * AF*_i * BF*_j  (computed on the fly)
// f32 WMMA 16x16x4: A = 16x4 tile of W, B = c-chunk replicated over 16 cols.
// W is streamed once (1.21 GB) -> non-temporal loads keep K/BF hot in L2.
// 4096 waves = 32 y-tiles x 128 k-slices; deterministic partials in ws.
// ---------------------------------------------------------------------------
__global__ void __launch_bounds__(GTHREADS)
gemv_wmma_kernel(const float* __restrict__ W, const float* __restrict__ K,
                 float* __restrict__ ws) {
  const int wv    = blockIdx.x * (GTHREADS / 32) + (threadIdx.x >> 5); // 0..4095
  const int lane  = threadIdx.x & 31;
  const int half  = lane >> 4;       // 0: K=0,1  1: K=2,3  (A & B layouts)
  const int m     = lane & 15;       // A-matrix row within tile
  const int ytile = wv & 31;         // 32 tiles of 16 outputs
  const int ks    = wv >> 5;         // 0..127 k-slices
  const int i0    = ks * IPERS;      // first K-row of this slice (KLEN = 6*768)

  const float* AFs = ws + WS_AFS;
  const float* BFs = ws + WS_BFS;
  const float* Wr0 = W + (size_t)(ytile * 16 + m) * (size_t)N2 + 2 * half;

  v8f acc0 = {0.f, 0.f, 0.f, 0.f, 0.f, 0.f, 0.f, 0.f};
  v8f acc1 = acc0;

  for (int ib = 0; ib < IPERS; ++ib) {
    const int i = i0 + ib;
    const float af = AFs[i];
    const float* Wr = Wr0 + (size_t)i * NAB;
    const float* Kr = K + (size_t)i * NAB + 2 * half;
    const float* Br = BFs + 2 * half;
    // warm the next row-chunk (3 KB/lane stride jump); speculative & safe.
    if (ib + 1 < IPERS)
      __builtin_prefetch(Wr + NAB, 0, 1);
#pragma unroll 2
    for (int jj = 0; jj < NAB; jj += 8) {
      v2f a0  = __builtin_nontemporal_load((const v2f*)(Wr + jj));
      v2f k0  = *(const v2f*)(Kr + jj);
      v2f bf0 = *(const v2f*)(Br + jj);
      v2f b0;
      b0.x = k0.x * af * bf0.x;
      b0.y = k0.y * af * bf0.y;
      acc0 = __builtin_amdgcn_wmma_f32_16x16x4_f32(false, a0, false, b0,
                                                   (short)0, acc0, false, false);
      v2f a1  = __builtin_nontemporal_load((const v2f*)(Wr + jj + 4));
      v2f k1  = *(const v2f*)(Kr + jj + 4);
      v2f bf1 = *(const v2f*)(Br + jj + 4);
      v2f b1;
      b1.x = k1.x * af * bf1.x;
      b1.y = k1.y * af * bf1.y;
      acc1 = __builtin_amdgcn_wmma_f32_16x16x4_f32(false, a1, false, b1,
                                                   (short)0, acc1, false, false);
    }
  }

  v8f acc = acc0 + acc1;
  // D layout: lane 0 holds M=0..7 (N=0), lane 16 holds M=8..15 (N=0);
  // all N columns are identical since B columns are replicated.
  if (m == 0) {
    float* part = ws + WS_PART + (size_t)ks * NY + ytile * 16 + half * 8;
    *(v8f*)part = acc;
  }
}

// ---------------------------------------------------------------------------
// Fixed-order reduction of 128 partials per output + bias (deterministic).
// ---------------------------------------------------------------------------
__global__ void reduce_kernel(const float* __restrict__ bias,
                              const float* __restrict__ ws,
                              float* __restrict__ Y) {
  int y = blockIdx.x * blockDim.x + threadIdx.x;
  if (y < NY) {
    float s = bias[y];
    const float* part = ws + WS_PART + y;
    for (int k = 0; k < KS; ++k) s += part[(size_t)k * NY];
    Y[y] = s;
  }
}

// ---------------------------------------------------------------------------
extern "C" void kernel_launch(void* const* d_in, const int* in_sizes, int n_in,
                              void* d_out, int out_size, void* d_ws, size_t ws_size,
                              hipStream_t stream) {
  (void)in_sizes; (void)n_in; (void)out_size; (void)ws_size;
  const float* AT = (const float*)d_in[0];
  const float* BT = (const float*)d_in[1];
  const float* K  = (const float*)d_in[2];
  const float* W  = (const float*)d_in[3];
  const float* b  = (const float*)d_in[4];
  float* Y  = (float*)d_out;
  float* ws = (float*)d_ws;

  init_kernel<<<1, 1024, 0, stream>>>(AT, BT, ws);
  solve_kernel<<<SBLK, STHREADS, 0, stream>>>(AT, BT, K, ws);
  gemv_wmma_kernel<<<GBLK, GTHREADS, 0, stream>>>(W, K, ws);
  reduce_kernel<<<(NY + 255) / 256, 256, 0, stream>>>(b, ws, Y);
}